// ContrastiveLoss_84293028151332
// MI455X (gfx1250) — compile-verified
//
#include <hip/hip_runtime.h>
#include <hip/hip_bf16.h>

// Supervised contrastive loss, MI455X (gfx1250), wave32 + f32 WMMA.
//
// sim = (E @ E^T) / 0.07 computed tile-wise with V_WMMA_F32_16X16X4_F32.
// One-pass flash-style online softmax per row; deterministic (no fp atomics).
// M-blocking: each wave carries TWO 16-row accumulators so every B fragment
// loaded from L2 is reused twice (halves L2 traffic vs 16 rows/workgroup).
//
// WMMA fragment layout (CDNA5 ISA 7.12.2, 32-bit operands, wave32):
//   A (16x4, MxK): VGPR0 = K {0 | 2} for lanes {0-15 | 16-31}, VGPR1 = K {1 | 3}; M = lane&15
//   B (4x16, KxN): mirrors A with N = lane&15
//   C (16x16):     VGPR j: lanes 0-15 -> (M=j, N=lane), lanes 16-31 -> (M=j+8, N=lane-16)

typedef __attribute__((ext_vector_type(2))) float v2f;
typedef __attribute__((ext_vector_type(8))) float v8f;

#define INV_TEMP 14.285714285714285f /* 1 / 0.07 */
#define ROWS_PER_WG 32               /* two 16-row subtiles per wave */

__global__ __launch_bounds__(256) void supcon_rows_kernel(
    const float* __restrict__ emb, const int* __restrict__ labels,
    float* __restrict__ row_loss, float* __restrict__ row_valid,
    int Bn, int Dn)
{
    // A panel: 32 rows x D (<=512) fp32, stride 516 so that 16 lanes reading the
    // same K at rows 0..15 hit 16 distinct LDS banks (516 mod 64 == 4). 66 KB.
    __shared__ float As[ROWS_PER_WG][516];
    __shared__ float sM[8][ROWS_PER_WG], sL[8][ROWS_PER_WG],
                     sP[8][ROWS_PER_WG], sC[8][ROWS_PER_WG];

    const int tid  = threadIdx.x;
    const int wave = tid >> 5;
    const int lane = tid & 31;
    const int half = lane >> 4;   // 0: lanes 0-15, 1: lanes 16-31
    const int ln   = lane & 15;
    const int r0   = blockIdx.x * ROWS_PER_WG;

    // Cooperative stage of the 32-row A panel into LDS (float4 vectors).
    const int d4 = Dn >> 2;
    for (int i = tid; i < ROWS_PER_WG * d4; i += 256) {
        const int row = i / d4;
        const int c   = (i - row * d4) << 2;
        const float4 v = *reinterpret_cast<const float4*>(emb + (size_t)(r0 + row) * Dn + c);
        *reinterpret_cast<float4*>(&As[row][c]) = v;
    }
    __syncthreads();

    // Row labels for the 2x8 rows this half-wave sees in the C layout.
    int lr[2][8];
#pragma unroll
    for (int s = 0; s < 2; ++s)
#pragma unroll
        for (int j = 0; j < 8; ++j) lr[s][j] = labels[r0 + s * 16 + j + 8 * half];

    // Online-softmax running state per row (replicated across the 16-lane half).
    float m[2][8], ls[2][8], ps[2][8], pc[2][8];
#pragma unroll
    for (int s = 0; s < 2; ++s)
#pragma unroll
        for (int j = 0; j < 8; ++j) {
            m[s][j] = -__builtin_inff();
            ls[s][j] = 0.f; ps[s][j] = 0.f; pc[s][j] = 0.f;
        }

    const int nTiles = Bn >> 4;   // 256 column tiles of width 16
    for (int ct = wave; ct < nTiles; ct += 8) {
        const int c0 = ct << 4;
        const float* __restrict__ Bp = emb + (size_t)(c0 + ln) * Dn; // column (c0+ln) of E^T
        if (ct + 8 < nTiles)  // hint next B panel for this wave -> global_prefetch_b8
            __builtin_prefetch(emb + (size_t)(c0 + 128 + ln) * Dn, 0, 1);

        v8f acc0 = {};
        v8f acc1 = {};
        for (int k = 0; k < Dn; k += 4) {
            const int ko = k + 2 * half;
            const v2f b  = *reinterpret_cast<const v2f*>(&Bp[ko]);         // 1 L2 load ...
            const v2f a0 = *reinterpret_cast<const v2f*>(&As[ln][ko]);     // ... reused by
            const v2f a1 = *reinterpret_cast<const v2f*>(&As[16 + ln][ko]);//     2 WMMAs
            acc0 = __builtin_amdgcn_wmma_f32_16x16x4_f32(
                false, a0, false, b, (short)0, acc0, false, false);
            acc1 = __builtin_amdgcn_wmma_f32_16x16x4_f32(
                false, a1, false, b, (short)0, acc1, false, false);
        }

        const int lc  = labels[c0 + ln];
        const int col = c0 + ln;

#pragma unroll
        for (int s = 0; s < 2; ++s) {
#pragma unroll
            for (int j = 0; j < 8; ++j) {
                const int row = r0 + s * 16 + j + 8 * half;
                const float v = (s == 0) ? acc0[j] : acc1[j];
                const float sim = v * INV_TEMP;

                // Tile row-max across the 16 lanes of this half (includes the
                // diagonal, matching the reference: max taken BEFORE masking).
                float tm = sim;
#pragma unroll
                for (int d = 1; d <= 8; d <<= 1) tm = fmaxf(tm, __shfl_xor(tm, d, 32));

                const float mOld = m[s][j];
                const float mNew = fmaxf(mOld, tm);

                float e = (col == row) ? 0.f : __expf(sim - mNew); // diag excluded from sums
                const bool pos = (lr[s][j] == lc) && (col != row);
                float p  = pos ? e : 0.f;
                float cn = pos ? 1.f : 0.f;
#pragma unroll
                for (int d = 1; d <= 8; d <<= 1) {
                    e  += __shfl_xor(e,  d, 32);
                    p  += __shfl_xor(p,  d, 32);
                    cn += __shfl_xor(cn, d, 32);
                }

                const float rescale = __expf(mOld - mNew);  // exp(-inf)=0 on first tile
                ls[s][j] = ls[s][j] * rescale + e;
                ps[s][j] = ps[s][j] * rescale + p;
                pc[s][j] += cn;
                m[s][j]  = mNew;
            }
        }
    }

    // Publish per-wave stats (all 16 lanes of a half hold identical values).
    if (ln == 0) {
#pragma unroll
        for (int s = 0; s < 2; ++s)
#pragma unroll
            for (int j = 0; j < 8; ++j) {
                const int rl = s * 16 + j + 8 * half;
                sM[wave][rl] = m[s][j];
                sL[wave][rl] = ls[s][j];
                sP[wave][rl] = ps[s][j];
                sC[wave][rl] = pc[s][j];
            }
    }
    __syncthreads();

    // Deterministic fixed-order 8-way merge, one thread per row.
    if (tid < ROWS_PER_WG) {
        float M = -__builtin_inff();
        for (int w = 0; w < 8; ++w) M = fmaxf(M, sM[w][tid]);
        float L = 0.f, P = 0.f, C = 0.f;
        for (int w = 0; w < 8; ++w) {
            const float f = __expf(sM[w][tid] - M);
            L += sL[w][tid] * f;
            P += sP[w][tid] * f;
            C += sC[w][tid];
        }
        const bool has = C > 0.5f;
        // -log(pos/all) = log(all) - log(pos); rows without positives contribute 0.
        row_loss[r0 + tid]  = has ? (logf(L) - logf(P)) : 0.f;
        row_valid[r0 + tid] = has ? 1.f : 0.f;
    }
}

__global__ __launch_bounds__(256) void supcon_final_kernel(
    const float* __restrict__ row_loss, const float* __restrict__ row_valid,
    float* __restrict__ out, int Bn)
{
    __shared__ float sl[256], sv[256];
    const int tid = threadIdx.x;
    float a = 0.f, n = 0.f;
    for (int i = tid; i < Bn; i += 256) { a += row_loss[i]; n += row_valid[i]; }
    sl[tid] = a; sv[tid] = n;
    __syncthreads();
    for (int s = 128; s > 0; s >>= 1) {
        if (tid < s) { sl[tid] += sl[tid + s]; sv[tid] += sv[tid + s]; }
        __syncthreads();
    }
    if (tid == 0)
        out[0] = (sv[0] > 0.f) ? (sl[0] / fmaxf(sv[0], 1.f)) : 0.f;
}

extern "C" void kernel_launch(void* const* d_in, const int* in_sizes, int n_in,
                              void* d_out, int out_size, void* d_ws, size_t ws_size,
                              hipStream_t stream) {
    const int Bn = in_sizes[1];              // 4096
    const int Dn = in_sizes[0] / Bn;         // 512
    const float* emb    = (const float*)d_in[0];
    const int*   labels = (const int*)d_in[1];

    float* row_loss  = (float*)d_ws;         // [Bn]
    float* row_valid = row_loss + Bn;        // [Bn]

    supcon_rows_kernel<<<Bn / ROWS_PER_WG, 256, 0, stream>>>(emb, labels, row_loss, row_valid, Bn, Dn);
    supcon_final_kernel<<<1, 256, 0, stream>>>(row_loss, row_valid, (float*)d_out, Bn);
}